// GCN_69733089018296
// MI455X (gfx1250) — compile-verified
//
#include <hip/hip_runtime.h>
#include <hip/hip_bf16.h>

// ---------------------------------------------------------------------------
// GCN (2-layer, PyG-style GCNConv + repulse + log_softmax) for MI455X gfx1250.
//  - GEMMs use V_WMMA_F32_16X16X4_F32 (full fp32 matrix core path).
//  - GEMM1 streams x (205 MB, the only HBM-bound operand) through the Tensor
//    Data Mover into double-buffered per-wave LDS tiles, hidden by TENSORcnt.
//  - Edge aggregation uses global f32 atomics into L2-resident buffers.
// ---------------------------------------------------------------------------

typedef __attribute__((ext_vector_type(2))) float v2f;
typedef __attribute__((ext_vector_type(8))) float v8f;
typedef __attribute__((ext_vector_type(4))) unsigned int u32x4;
typedef __attribute__((ext_vector_type(8))) int i32x8;
typedef __attribute__((ext_vector_type(4))) int i32x4;

#define NN      100000    // nodes
#define NE      3200000   // edges
#define INF_    512       // input features
#define HID     16        // hidden features
#define NCLS    40        // classes
#define NCLS_P  48        // padded to 3x16 WMMA tiles
#define KC      64        // GEMM1 K-chunk staged in LDS per TDM transfer

#define M_TILES (NN / 16) // 6250, exact

#if __has_builtin(__builtin_amdgcn_tensor_load_to_lds)
#define HAVE_TDM 1
#endif

// ---------------- degree / norm precompute ----------------

__global__ void k_init_deg(float* deg1, float* degr) {
    unsigned i = blockIdx.x * blockDim.x + threadIdx.x;
    if (i < NN) { deg1[i] = 1.0f; /* self loop */ degr[i] = 0.0f; }
}

__global__ void k_deg_edges(const int* __restrict__ dst, float* deg1) {
    unsigned i = blockIdx.x * blockDim.x + threadIdx.x;
    if (i < NE) atomicAdd(&deg1[dst[i]], 1.0f);
}

__global__ void k_deg_rep(const int* __restrict__ rcol, float* degr) {
    unsigned i = blockIdx.x * blockDim.x + threadIdx.x;
    if (i < NN) atomicAdd(&degr[rcol[i]], 1.0f);
}

__global__ void k_dinv(float* deg1, float* degr) {
    unsigned i = blockIdx.x * blockDim.x + threadIdx.x;
    if (i >= NN) return;
    float d1 = deg1[i];                 // >= 1 always (self loop)
    deg1[i] = rsqrtf(d1);
    float dr = degr[i];
    degr[i] = (dr > 0.0f) ? rsqrtf(dr) : 0.0f;
}

// ---------------- TDM helpers ----------------

__device__ __forceinline__ unsigned lds_offset_of(const void* p) {
    // generic -> LDS addrspace -> 32-bit LDS byte offset
    return (unsigned)(unsigned long long)(__attribute__((address_space(3))) const void*)p;
}

#ifdef HAVE_TDM
// DMA a 16-row x KC-float fp32 tile (row stride INF_ floats) into LDS.
// D# encoding per CDNA5 ISA (2D: upper groups zero).
// 6-arg builtin form (clang-23 / therock headers): (g0,g1,g2,g3,g4,cpol).
__device__ __forceinline__ void tdm_load_tile(unsigned lds_byte, const float* gsrc) {
    unsigned long long ga = (unsigned long long)(const void*)gsrc;
    unsigned ga_lo = (unsigned)__builtin_amdgcn_readfirstlane((int)(unsigned)ga);
    unsigned ga_hi = (unsigned)__builtin_amdgcn_readfirstlane((int)(unsigned)(ga >> 32));
    unsigned lb    = (unsigned)__builtin_amdgcn_readfirstlane((int)lds_byte);

    u32x4 g0;
    g0.x = 1u;                                    // count=1 valid descriptor
    g0.y = lb;                                    // lds_addr  (bits 63:32)
    g0.z = ga_lo;                                 // global_addr[31:0]
    g0.w = (ga_hi & 0x01FFFFFFu) | (2u << 30);    // global_addr[56:32] | type=2

    i32x8 g1;
    g1[0] = 0x00020000;          // workgroup_mask=0, data_size=2 (4 bytes)
    g1[1] = (int)(INF_ << 16);   // tensor_dim0[15:0]=512 in bits[63:48]
    g1[2] = (int)(16 << 16);     // tensor_dim1[15:0]=16 in bits[95:80]
    g1[3] = (int)(KC << 16);     // tile_dim0=64 in bits[127:112]
    g1[4] = 16;                  // tile_dim1=16, tile_dim2=0
    g1[5] = INF_;                // tensor_dim0_stride[31:0]=512
    g1[6] = 0;
    g1[7] = 0;

    i32x4 gz4 = {0, 0, 0, 0};                     // 2D: groups 2/3 unused
    i32x8 gz8 = {0, 0, 0, 0, 0, 0, 0, 0};
    __builtin_amdgcn_tensor_load_to_lds(g0, g1, gz4, gz4, gz8, 0);
}
#endif

// ---------------- GEMM1: xw = x @ W1  (100000x512 @ 512x16), fp32 WMMA -----
// 128 threads = 4 waves; one 16-row M tile per wave; K in 8 chunks of 64
// staged into a per-wave double-buffered LDS slice (TDM path) or via
// coalesced float4 loads + ds_store_b128 (fallback).

__global__ void __launch_bounds__(128)
k_gemm1_wmma(const float* __restrict__ x, const float* __restrict__ W1,
             float* __restrict__ xw) {
    __shared__ float ldsW1[INF_ * HID];           // 32 KB
    __shared__ float ldsA[4][2][16 * KC];         // 4 waves x 2 bufs x 4 KB
    for (int t = threadIdx.x; t < (INF_ * HID) / 4; t += blockDim.x)
        ((float4*)ldsW1)[t] = ((const float4*)W1)[t];
    __syncthreads();

    const int lane  = threadIdx.x & 31;
    const int wid   = threadIdx.x >> 5;
    const int tile  = blockIdx.x * 4 + wid;       // 16-row tile of M
    if (tile >= M_TILES) return;
    const int row0  = tile * 16;

    // A 16x4 f32 frag: lanes 0-15 hold M=lane,K={0,1}; lanes 16-31 M,K={2,3}
    const int m     = lane & 15;
    const int kbase = (lane >> 4) * 2;
    const int n     = lane & 15;

    float* ab0 = &ldsA[wid][0][0];
    float* ab1 = &ldsA[wid][1][0];
    const float* xbase = x + (size_t)row0 * INF_;

    v8f c = {};

#ifdef HAVE_TDM
    const unsigned lo0 = lds_offset_of(ab0);
    const unsigned lo1 = lds_offset_of(ab1);
    tdm_load_tile(lo0, xbase);                    // chunk 0 in flight
    for (int chunk = 0; chunk < INF_ / KC; ++chunk) {
        if (chunk + 1 < INF_ / KC) {
            tdm_load_tile((chunk & 1) ? lo0 : lo1, xbase + (chunk + 1) * KC);
            __builtin_amdgcn_s_wait_tensorcnt(1); // current chunk landed
        } else {
            __builtin_amdgcn_s_wait_tensorcnt(0);
        }
        __asm__ volatile("" ::: "memory");        // keep LDS reads below wait
        const float* ab = (chunk & 1) ? ab1 : ab0;
        const int kc = chunk * KC;
        #pragma unroll
        for (int kk = 0; kk < KC; kk += 4) {
            v2f a, b;
            a.x = ab[m * KC + kk + kbase];
            a.y = ab[m * KC + kk + kbase + 1];
            b.x = ldsW1[(kc + kk + kbase) * HID + n];
            b.y = ldsW1[(kc + kk + kbase + 1) * HID + n];
            c = __builtin_amdgcn_wmma_f32_16x16x4_f32(
                    false, a, false, b, (short)0, c, false, false);
        }
    }
#else
    for (int chunk = 0; chunk < INF_ / KC; ++chunk) {
        const int kc = chunk * KC;
        // coalesced stage: 16 rows x 64 floats = 256 float4, 8 per lane;
        // consecutive lanes hit consecutive 16B chunks of one row.
        #pragma unroll
        for (int cidx = lane; cidx < 16 * (KC / 4); cidx += 32) {
            int row = cidx >> 4, c4 = (cidx & 15) * 4;
            *(float4*)&ab0[row * KC + c4] =
                *(const float4*)(xbase + (size_t)row * INF_ + kc + c4);
        }
        if (chunk + 1 < INF_ / KC)
            __builtin_prefetch(xbase + (size_t)m * INF_ + kc + KC, 0, 1);
        // same-wave LDS ops are in-order: stores above land before loads below
        #pragma unroll
        for (int kk = 0; kk < KC; kk += 4) {
            v2f a, b;
            a.x = ab0[m * KC + kk + kbase];
            a.y = ab0[m * KC + kk + kbase + 1];
            b.x = ldsW1[(kc + kk + kbase) * HID + n];
            b.y = ldsW1[(kc + kk + kbase + 1) * HID + n];
            c = __builtin_amdgcn_wmma_f32_16x16x4_f32(
                    false, a, false, b, (short)0, c, false, false);
        }
    }
    (void)ab1;
#endif

    // D 16x16 f32: VGPR r -> row (r + 8*(lane>=16)), col lane&15
    const int rbase = (lane >> 4) * 8;
    #pragma unroll
    for (int r = 0; r < 8; ++r)
        xw[(size_t)(row0 + rbase + r) * HID + n] = c[r];
}

// ---------------- layer-1 aggregation ----------------

__global__ void k_init_agg1(const float* __restrict__ xw,
                            const float* __restrict__ dinv1,
                            const float* __restrict__ b1,
                            float* __restrict__ agg1) {
    unsigned i = blockIdx.x * blockDim.x + threadIdx.x;
    if (i >= NN * HID) return;
    unsigned node = i >> 4, f = i & 15;
    float di = dinv1[node];
    agg1[i] = b1[f] + di * di * xw[i];            // bias + self-loop term
}

__global__ void k_agg1_edges(const int* __restrict__ src,
                             const int* __restrict__ dst,
                             const float* __restrict__ dinv1,
                             const float* __restrict__ xw,
                             float* __restrict__ agg1) {
    unsigned gid = blockIdx.x * blockDim.x + threadIdx.x;
    if (gid >= (unsigned)NE * HID) return;
    unsigned e = gid >> 4, f = gid & 15;
    int s = src[e], d = dst[e];
    float w = dinv1[s] * dinv1[d];
    atomicAdd(&agg1[(size_t)d * HID + f], w * xw[(size_t)s * HID + f]);
}

__global__ void k_repulse1_relu(const float* __restrict__ agg1,
                                const int* __restrict__ rrow,
                                const int* __restrict__ rcol,
                                const float* __restrict__ dinvr,
                                float* __restrict__ h1) {
    unsigned i = blockIdx.x * blockDim.x + threadIdx.x;
    if (i >= NN * HID) return;
    unsigned node = i >> 4, f = i & 15;
    int rc = rcol[node];
    float nr = dinvr[rrow[node]] * dinvr[rc];
    float v = agg1[i] - nr * agg1[(size_t)rc * HID + f];
    h1[i] = fmaxf(v, 0.0f);
}

// ---------------- GEMM2: hw2 = h1 @ W2 (100000x16 @ 16x40), fp32 WMMA -----

__global__ void __launch_bounds__(256)
k_gemm2_wmma(const float* __restrict__ h1, const float* __restrict__ W2,
             float* __restrict__ hw2) {
    __shared__ float ldsW2[HID * NCLS_P];         // 16x48, zero padded
    for (int t = threadIdx.x; t < HID * NCLS_P; t += blockDim.x)
        ldsW2[t] = 0.0f;
    __syncthreads();
    for (int t = threadIdx.x; t < HID * NCLS; t += blockDim.x)
        ldsW2[(t / NCLS) * NCLS_P + (t % NCLS)] = W2[t];
    __syncthreads();

    const int lane  = threadIdx.x & 31;
    const int wid   = threadIdx.x >> 5;
    const int tile  = blockIdx.x * 8 + wid;
    if (tile >= M_TILES) return;
    const int row0  = tile * 16;

    const int m     = lane & 15;
    const int kbase = (lane >> 4) * 2;
    const int n     = lane & 15;
    const float* hrow = h1 + (size_t)(row0 + m) * HID;

    v8f c0 = {}, c1 = {}, c2 = {};
    #pragma unroll
    for (int k0 = 0; k0 < HID; k0 += 4) {
        v2f a;
        a.x = hrow[k0 + kbase];
        a.y = hrow[k0 + kbase + 1];
        const float* w0 = &ldsW2[(k0 + kbase) * NCLS_P];
        const float* w1 = &ldsW2[(k0 + kbase + 1) * NCLS_P];
        v2f b;
        b.x = w0[n];       b.y = w1[n];
        c0 = __builtin_amdgcn_wmma_f32_16x16x4_f32(false, a, false, b, (short)0, c0, false, false);
        b.x = w0[16 + n];  b.y = w1[16 + n];
        c1 = __builtin_amdgcn_wmma_f32_16x16x4_f32(false, a, false, b, (short)0, c1, false, false);
        b.x = w0[32 + n];  b.y = w1[32 + n];
        c2 = __builtin_amdgcn_wmma_f32_16x16x4_f32(false, a, false, b, (short)0, c2, false, false);
    }

    const int rbase = (lane >> 4) * 8;
    #pragma unroll
    for (int r = 0; r < 8; ++r) {
        size_t row = (size_t)(row0 + rbase + r);
        hw2[row * NCLS + n]      = c0[r];
        hw2[row * NCLS + 16 + n] = c1[r];
        if (32 + n < NCLS)
            hw2[row * NCLS + 32 + n] = c2[r];
    }
}

// ---------------- layer-2 aggregation ----------------

__global__ void k_init_agg2(const float* __restrict__ hw2,
                            const float* __restrict__ dinv1,
                            const float* __restrict__ b2,
                            float* __restrict__ agg2) {
    unsigned i = blockIdx.x * blockDim.x + threadIdx.x;
    if (i >= (unsigned)NN * NCLS) return;
    unsigned node = i / NCLS, f = i % NCLS;
    float di = dinv1[node];
    agg2[i] = b2[f] + di * di * hw2[i];
}

__global__ void k_agg2_edges(const int* __restrict__ src,
                             const int* __restrict__ dst,
                             const float* __restrict__ dinv1,
                             const float* __restrict__ hw2,
                             float* __restrict__ agg2) {
    unsigned gid = blockIdx.x * blockDim.x + threadIdx.x;
    if (gid >= (unsigned)NE * NCLS) return;
    unsigned e = gid / NCLS, f = gid % NCLS;
    int s = src[e], d = dst[e];
    float w = dinv1[s] * dinv1[d];
    atomicAdd(&agg2[(size_t)d * NCLS + f], w * hw2[(size_t)s * NCLS + f]);
}

__global__ void k_repulse2_logsoftmax(const float* __restrict__ agg2,
                                      const int* __restrict__ rrow,
                                      const int* __restrict__ rcol,
                                      const float* __restrict__ dinvr,
                                      float* __restrict__ out) {
    unsigned node = blockIdx.x * blockDim.x + threadIdx.x;
    if (node >= NN) return;
    int rc = rcol[node];
    float nr = dinvr[rrow[node]] * dinvr[rc];
    const float* a = agg2 + (size_t)node * NCLS;
    const float* b = agg2 + (size_t)rc * NCLS;

    float v[NCLS];
    float m = -3.4e38f;
    #pragma unroll
    for (int f = 0; f < NCLS; ++f) {
        float t = a[f] - nr * b[f];
        v[f] = t;
        m = fmaxf(m, t);
    }
    float s = 0.0f;
    #pragma unroll
    for (int f = 0; f < NCLS; ++f) s += __expf(v[f] - m);
    float lse = __logf(s);
    float* o = out + (size_t)node * NCLS;
    #pragma unroll
    for (int f = 0; f < NCLS; ++f) o[f] = v[f] - m - lse;
}

// ---------------- host launcher ----------------

extern "C" void kernel_launch(void* const* d_in, const int* in_sizes, int n_in,
                              void* d_out, int out_size, void* d_ws, size_t ws_size,
                              hipStream_t stream) {
    const float* x    = (const float*)d_in[0];
    const int*   ei   = (const int*)d_in[1];      // [2, NE]
    const int*   rei  = (const int*)d_in[2];      // [2, NN]
    const float* W1   = (const float*)d_in[3];
    const float* b1   = (const float*)d_in[4];
    const float* W2   = (const float*)d_in[5];
    const float* b2   = (const float*)d_in[6];
    float* out = (float*)d_out;

    const int* src  = ei;
    const int* dst  = ei + NE;
    const int* rrow = rei;
    const int* rcol = rei + NN;

    // workspace layout (floats), 16-float aligned regions; total ~52 MB
    float* ws    = (float*)d_ws;
    float* dinv1 = ws;                         // NN      (deg1 -> dinv1 in place)
    float* dinvr = ws + 100352;                // NN      (degr -> dinvr in place)
    float* xw    = ws + 200704;                // NN*HID
    float* agg1  = ws + 1800704;               // NN*HID
    float* h1    = ws + 3400704;               // NN*HID
    float* hw2   = ws + 5000704;               // NN*NCLS
    float* agg2  = ws + 9000704;               // NN*NCLS

    const int B = 256;
    // degrees / norms
    k_init_deg <<<(NN + B - 1) / B, B, 0, stream>>>(dinv1, dinvr);
    k_deg_edges<<<(NE + B - 1) / B, B, 0, stream>>>(dst, dinv1);
    k_deg_rep  <<<(NN + B - 1) / B, B, 0, stream>>>(rcol, dinvr);
    k_dinv     <<<(NN + B - 1) / B, B, 0, stream>>>(dinv1, dinvr);

    // layer 1
    k_gemm1_wmma<<<(M_TILES + 3) / 4, 128, 0, stream>>>(x, W1, xw);
    k_init_agg1 <<<(NN * HID + B - 1) / B, B, 0, stream>>>(xw, dinv1, b1, agg1);
    k_agg1_edges<<<((unsigned)NE * HID + B - 1) / B, B, 0, stream>>>(src, dst, dinv1, xw, agg1);
    k_repulse1_relu<<<(NN * HID + B - 1) / B, B, 0, stream>>>(agg1, rrow, rcol, dinvr, h1);

    // layer 2
    k_gemm2_wmma<<<(M_TILES + 7) / 8, B, 0, stream>>>(h1, W2, hw2);
    k_init_agg2 <<<((unsigned)NN * NCLS + B - 1) / B, B, 0, stream>>>(hw2, dinv1, b2, agg2);
    k_agg2_edges<<<((unsigned)NE * NCLS + B - 1) / B, B, 0, stream>>>(src, dst, dinv1, hw2, agg2);
    k_repulse2_logsoftmax<<<(NN + B - 1) / B, B, 0, stream>>>(agg2, rrow, rcol, dinvr, out);
}